// VirtualNormal_71949292142903
// MI455X (gfx1250) — compile-verified
//
#include <hip/hip_runtime.h>

// Problem constants (match reference)
#define NB 8
#define NC 3
#define NH 768
#define NW 1024
#define TPB 256

#define DCOS  0.867f
#define DDIFF 0.005f
#define DZ    0.0001f

#if __has_builtin(__builtin_amdgcn_global_load_async_to_lds_b32)
#define HAS_ASYNC 1
#else
#define HAS_ASYNC 0
#endif

typedef __attribute__((address_space(1))) int gas_int;
typedef __attribute__((address_space(3))) int las_int;

__device__ __forceinline__ void stage_b32(const int* g, int* l) {
#if HAS_ASYNC
  // gfx1250 async global->LDS copy (ASYNCcnt-tracked); per-lane LDS dest addr.
  __builtin_amdgcn_global_load_async_to_lds_b32(
      (gas_int*)const_cast<int*>(g),
      (las_int*)l,
      /*offset=*/0, /*cpol=*/0);
#else
  *l = *g;
#endif
}

__device__ __forceinline__ void wait_async0() {
#if HAS_ASYNC
#if __has_builtin(__builtin_amdgcn_s_wait_asynccnt)
  __builtin_amdgcn_s_wait_asynccnt(0);
  asm volatile("" ::: "memory");
#else
  asm volatile("s_wait_asynccnt 0" ::: "memory");
#endif
#endif
}

__global__ void __launch_bounds__(TPB)
vn_partial_kernel(const float* __restrict__ gt, const float* __restrict__ pr,
                  const int* __restrict__ p1x, const int* __restrict__ p1y,
                  const int* __restrict__ p2x, const int* __restrict__ p2y,
                  const int* __restrict__ p3x, const int* __restrict__ p3y,
                  int G, float2* __restrict__ partial) {
  __shared__ int   sidx[6][TPB];
  __shared__ float snum[TPB];
  __shared__ float sden[TPB];

  const int  t     = threadIdx.x;
  const int  g     = blockIdx.x * TPB + t;
  const int  b     = blockIdx.y;
  const bool valid = (g < G);
  const int  gc    = valid ? g : (G - 1);

  // Stage the six coalesced index streams through LDS via async copies.
  stage_b32(p1x + gc, &sidx[0][t]);
  stage_b32(p1y + gc, &sidx[1][t]);
  stage_b32(p2x + gc, &sidx[2][t]);
  stage_b32(p2y + gc, &sidx[3][t]);
  stage_b32(p3x + gc, &sidx[4][t]);
  stage_b32(p3y + gc, &sidx[5][t]);
  wait_async0();

  const int x1 = sidx[0][t], y1 = sidx[1][t];
  const int x2 = sidx[2][t], y2 = sidx[3][t];
  const int x3 = sidx[4][t], y3 = sidx[5][t];

  const int o1 = y1 * NW + x1;
  const int o2 = y2 * NW + x2;
  const int o3 = y3 * NW + x3;
  const int basec = b * NC * NH * NW;

  // A[c][p]: gt groups, P[c][p]: pred groups (c = channel, p = point)
  float A[3][3], P[3][3];
#pragma unroll
  for (int c = 0; c < 3; ++c) {
    const int cb = basec + c * (NH * NW);
    A[c][0] = gt[cb + o1]; A[c][1] = gt[cb + o2]; A[c][2] = gt[cb + o3];
    P[c][0] = pr[cb + o1]; P[c][1] = pr[cb + o2]; P[c][2] = pr[cb + o3];
  }

  // pw_diff vectors over channels: d0 = p2-p1, d1 = p3-p1, d2 = p3-p2
  float d0[3], d1[3], d2[3];
#pragma unroll
  for (int c = 0; c < 3; ++c) {
    d0[c] = A[c][1] - A[c][0];
    d1[c] = A[c][2] - A[c][0];
    d2[c] = A[c][2] - A[c][1];
  }

  auto dot3 = [](const float* u, const float* v) {
    return u[0] * v[0] + u[1] * v[1] + u[2] * v[2];
  };
  const float e00 = dot3(d0, d0), e11 = dot3(d1, d1), e22 = dot3(d2, d2);
  const float e01 = dot3(d0, d1), e02 = dot3(d0, d2), e12 = dot3(d1, d2);
  const float n0 = sqrtf(e00), n1 = sqrtf(e11), n2 = sqrtf(e22);

  auto big = [](float num, float den) {
    const float ne = num / (den + 1e-8f);
    return (int)((ne > DCOS) || (ne < -DCOS));
  };
  // 3x3 norm_energy count includes diagonal and both symmetric off-diagonals
  const int cnt = big(e00, n0 * n0) + big(e11, n1 * n1) + big(e22, n2 * n2) +
                  2 * (big(e01, n0 * n1) + big(e02, n0 * n2) + big(e12, n1 * n2));
  const bool mask_cos = cnt > 3;
  const bool mask_pad = (A[2][0] > DZ) && (A[2][1] > DZ) && (A[2][2] > DZ);
  const bool mx = (fabsf(d0[0]) < DDIFF) || (fabsf(d1[0]) < DDIFF) || (fabsf(d2[0]) < DDIFF);
  const bool my = (fabsf(d0[1]) < DDIFF) || (fabsf(d1[1]) < DDIFF) || (fabsf(d2[1]) < DDIFF);
  const bool mz = (fabsf(d0[2]) < DDIFF) || (fabsf(d1[2]) < DDIFF) || (fabsf(d2[2]) < DDIFF);
  const bool mask = mask_pad && !((mx && my && mz) || mask_cos);

  // Replicate reference broadcast: zmask indexed by point p, applied to channel row p.
  const bool zm0 = (P[2][0] == 0.0f);
  const bool zm1 = (P[2][1] == 0.0f);
  const bool zm2 = (P[2][2] == 0.0f);
  if (zm0) { P[0][0] = 1e-4f; P[0][1] = 1e-4f; P[0][2] = 1e-4f; }
  if (zm1) { P[1][0] = 1e-4f; P[1][1] = 1e-4f; P[1][2] = 1e-4f; }
  if (zm2) { P[2][0] = 1e-4f; P[2][1] = 1e-4f; P[2][2] = 1e-4f; }

  // Normals: cross(p12, p13) over channel axis
  float gn[3], pn[3];
  gn[0] = d0[1] * d1[2] - d0[2] * d1[1];
  gn[1] = d0[2] * d1[0] - d0[0] * d1[2];
  gn[2] = d0[0] * d1[1] - d0[1] * d1[0];
  float gl = sqrtf(gn[0] * gn[0] + gn[1] * gn[1] + gn[2] * gn[2]);
  if (gl == 0.0f) gl = 0.01f;

  float q0[3], q1[3];
#pragma unroll
  for (int c = 0; c < 3; ++c) {
    q0[c] = P[c][1] - P[c][0];
    q1[c] = P[c][2] - P[c][0];
  }
  pn[0] = q0[1] * q1[2] - q0[2] * q1[1];
  pn[1] = q0[2] * q1[0] - q0[0] * q1[2];
  pn[2] = q0[0] * q1[1] - q0[1] * q1[0];
  float pl = sqrtf(pn[0] * pn[0] + pn[1] * pn[1] + pn[2] * pn[2]);
  if (pl == 0.0f) pl = 0.01f;

  const float pg = fabsf(gn[0] / gl - pn[0] / pl) +
                   fabsf(gn[1] / gl - pn[1] / pl) +
                   fabsf(gn[2] / gl - pn[2] / pl);

  const bool take = valid && mask;
  snum[t] = take ? pg : 0.0f;
  sden[t] = take ? 1.0f : 0.0f;
  __syncthreads();
#pragma unroll
  for (int s = TPB / 2; s > 0; s >>= 1) {
    if (t < s) { snum[t] += snum[t + s]; sden[t] += sden[t + s]; }
    __syncthreads();
  }
  if (t == 0) {
    partial[blockIdx.y * gridDim.x + blockIdx.x] = make_float2(snum[0], sden[0]);
  }
}

__global__ void __launch_bounds__(TPB)
vn_final_kernel(const float2* __restrict__ partial, int n, float* __restrict__ out) {
  __shared__ float sn[TPB];
  __shared__ float sd[TPB];
  float a = 0.0f, c = 0.0f;
  for (int i = threadIdx.x; i < n; i += TPB) {
    const float2 p = partial[i];
    a += p.x;
    c += p.y;
  }
  sn[threadIdx.x] = a;
  sd[threadIdx.x] = c;
  __syncthreads();
#pragma unroll
  for (int s = TPB / 2; s > 0; s >>= 1) {
    if (threadIdx.x < s) {
      sn[threadIdx.x] += sn[threadIdx.x + s];
      sd[threadIdx.x] += sd[threadIdx.x + s];
    }
    __syncthreads();
  }
  if (threadIdx.x == 0) out[0] = sn[0] / fmaxf(sd[0], 1.0f);
}

extern "C" void kernel_launch(void* const* d_in, const int* in_sizes, int n_in,
                              void* d_out, int out_size, void* d_ws, size_t ws_size,
                              hipStream_t stream) {
  const float* gt  = (const float*)d_in[0];
  const float* pr  = (const float*)d_in[1];
  const int*   p1x = (const int*)d_in[2];
  const int*   p1y = (const int*)d_in[3];
  const int*   p2x = (const int*)d_in[4];
  const int*   p2y = (const int*)d_in[5];
  const int*   p3x = (const int*)d_in[6];
  const int*   p3y = (const int*)d_in[7];

  const int G   = in_sizes[2];
  const int nbx = (G + TPB - 1) / TPB;

  float2* partial = (float2*)d_ws;

  dim3 grid(nbx, NB);
  vn_partial_kernel<<<grid, TPB, 0, stream>>>(gt, pr, p1x, p1y, p2x, p2y, p3x, p3y,
                                              G, partial);
  vn_final_kernel<<<1, TPB, 0, stream>>>(partial, nbx * NB, (float*)d_out);
}